// TopDownProjector_82806969467611
// MI455X (gfx1250) — compile-verified
//
#include <hip/hip_runtime.h>
#include <hip/hip_bf16.h>
#include <stdint.h>

// TopDownProjector for volume [B,C,H,W,D] = [2,1,512,512,256], f32, depth contiguous.
// One wave32 per column: lanes cover 32 consecutive depths starting from the top
// (base = D-32), one coalesced 128B non-temporal load per slab, ballot to find the
// highest non-zero depth, early-exit. Expected slabs touched per column ~= 1 for the
// random-label input (P(slab all-zero) = 9^-32), and the loop is exact for any input
// (all-zero column -> (0,0)). Hit value extracted with v_readlane_b32 (lane index is
// wave-uniform from the SGPR ballot mask), so no LDS is used at all.

#define TDP_D 256

__global__ __launch_bounds__(256) void topdown_proj_kernel(
    const float* __restrict__ vol,   // [numCols, D] contiguous in D
    float* __restrict__ height,      // [numCols]
    float* __restrict__ seg,         // [numCols]
    int numCols) {
  const int lane = threadIdx.x & 31;
  const int waveInBlock = threadIdx.x >> 5;
  const int wavesPerBlock = blockDim.x >> 5;
  const int col = blockIdx.x * wavesPerBlock + waveInBlock;
  if (col >= numCols) return;   // uniform per wave, no intra-wave divergence

  const float* colp = vol + (size_t)col * TDP_D;

  float hf = 0.0f;  // height_field (depth index of first hit from top), 0 if empty
  float sm = 0.0f;  // seg_map (label at hit), 0 if empty

#pragma unroll 1
  for (int base = TDP_D - 32; base >= 0; base -= 32) {
    // Coalesced 128B slab load; streamed once -> non-temporal (TH=NT).
    float v = __builtin_nontemporal_load(colp + base + lane);
    // wave32 ballot: bit L set iff lane L's voxel is non-zero (low 32 bits valid)
    unsigned mask = (unsigned)__ballot(v != 0.0f);
    if (mask) {
      // Highest set bit = largest depth index with a non-zero voxel in this slab;
      // slabs are scanned top-down, so this is the global top-down first hit.
      int hitLane = 31 - __clz((int)mask);   // wave-uniform (mask is an SGPR)
      hf = (float)(base + hitLane);
      sm = __int_as_float(__builtin_amdgcn_readlane(__float_as_int(v), hitLane));
      break;
    }
  }

  if (lane == 0) {
    height[col] = hf;
    seg[col] = sm;
  }
}

extern "C" void kernel_launch(void* const* d_in, const int* in_sizes, int n_in,
                              void* d_out, int out_size, void* d_ws, size_t ws_size,
                              hipStream_t stream) {
  (void)n_in; (void)d_ws; (void)ws_size; (void)out_size;

  const float* vol = (const float*)d_in[0];
  const int total = in_sizes[0];          // B*C*H*W*D = 134217728
  const int numCols = total / TDP_D;      // 524288 columns

  float* height = (float*)d_out;          // outputs concatenated: height_field, seg_map
  float* seg = height + numCols;

  const int block = 256;                  // 8 wave32s -> 8 columns per block
  const int wavesPerBlock = block / 32;
  const int grid = (numCols + wavesPerBlock - 1) / wavesPerBlock;  // 65536

  topdown_proj_kernel<<<grid, block, 0, stream>>>(vol, height, seg, numCols);
}